// NLRidge_63496796504803
// MI455X (gfx1250) — compile-verified
//
#include <hip/hip_runtime.h>
#include <hip/hip_bf16.h>

#define H_IMG 256
#define W_IMG 256
#define N_IMG 2
#define C_IMG 3
#define P_PATCH 7
#define WIN_SZ 37
#define V_HALF 18
#define NSITE 64
#define HOWO 250     // H - p + 1
#define HPMAX 249    // H - p
#define NFEAT 147    // C*p*p
#define NPAD 160     // NFEAT padded to multiple of 16
#define HW (H_IMG * W_IMG)
#define WINEXT 43    // WIN + p - 1

typedef __attribute__((ext_vector_type(2))) float v2f;
typedef __attribute__((ext_vector_type(8))) float v8f;

__device__ __forceinline__ v8f wmma_f32x4(v2f a, v2f b, v8f c) {
  // D(16x16,f32) = A(16x4,f32) * B(4x16,f32) + C
  return __builtin_amdgcn_wmma_f32_16x16x4_f32(false, a, false, b, (short)0, c,
                                               false, false);
}

// Direct L2 -> LDS copy (CDNA5 async path, tracked by ASYNCcnt).
// Low 32 bits of a generic pointer in the LDS aperture = wave LDS address.
__device__ __forceinline__ void async_g2lds_b32(void* lds_dst,
                                                const void* gsrc) {
  unsigned la = (unsigned)(unsigned long long)lds_dst;
  asm volatile("global_load_async_to_lds_b32 %0, %1, off"
               :
               : "v"(la), "v"(gsrc)
               : "memory");
}
__device__ __forceinline__ void wait_async0() {
  asm volatile("s_wait_asynccnt 0" ::: "memory");
}

// ---------------------------------------------------------------------------
// channel mean: yb[b, i, j] = mean_c y[b, c, i, j]
// ---------------------------------------------------------------------------
__global__ void mean_kernel(const float* __restrict__ y, float* __restrict__ out,
                            int n) {
  int i = blockIdx.x * blockDim.x + threadIdx.x;
  if (i >= n) return;
  int b = i / HW, pix = i % HW;
  const float* p = y + (long)b * C_IMG * HW + pix;
  out[i] = (p[0] + p[HW] + p[2 * HW]) * (1.0f / 3.0f);
}

__global__ void zero_kernel(float* __restrict__ p, int n) {
  int i = blockIdx.x * blockDim.x + threadIdx.x;
  if (i < n) p[i] = 0.0f;
}

// out = xsum / wmap (wmap broadcast over channels)
__global__ void divide_kernel(const float* __restrict__ xsum,
                              const float* __restrict__ wmap,
                              float* __restrict__ out, int n) {
  int i = blockIdx.x * blockDim.x + threadIdx.x;
  if (i >= n) return;
  int b = i / (C_IMG * HW);
  int pix = i % HW;
  out[i] = xsum[i] / wmap[b * HW + pix];
}

// ---------------------------------------------------------------------------
// Block matching. Replicates align_corners(inf) padded-coordinate semantics:
// padded coord q in [0,252]; 250/251 invalid; 249 valid off stride-4 grid;
// 252 maps to 249 on the stride-4 grid.
// ---------------------------------------------------------------------------
__device__ __forceinline__ bool mapq(int qr, int qc, int& ar, int& ac) {
  if (qr < 0 || qr > 252 || qr == 250 || qr == 251) return false;
  if (qc < 0 || qc > 252 || qc == 250 || qc == 251) return false;
  int tr = (qr <= 248) ? 0 : ((qr == 249) ? 1 : 2);
  int tc = (qc <= 248) ? 0 : ((qc == 249) ? 1 : 2);
  if (tr == 0 && tc == 0) { ar = qr; ac = qc; return true; }
  if (tr == 1) {  // row 249: stride-aligned columns were relocated -> inf
    if (tc == 0 && (qc & 3) != 0) { ar = 249; ac = qc; return true; }
    return false;
  }
  if (tr == 2) {  // row 252 holds row-249 values at stride-4 columns
    if (tc == 0 && (qc & 3) == 0) { ar = 249; ac = qc; return true; }
    if (tc == 2) { ar = 249; ac = 249; return true; }
    return false;
  }
  // tr == 0, tc != 0
  if (tc == 1) {
    if ((qr & 3) != 0) { ar = qr; ac = 249; return true; }
    return false;
  }
  // tc == 2
  if ((qr & 3) == 0) { ar = qr; ac = 249; return true; }
  return false;
}

template <int M>
__global__ void __launch_bounds__(64) bm_kernel(const float* __restrict__ meanimg,
                                                int* __restrict__ idxout) {
  __shared__ float win[WINEXT * WINEXT];
  __shared__ float dist[WIN_SZ * WIN_SZ];
  __shared__ float rbest[64];
  __shared__ int ribest[64];

  const int tid = threadIdx.x;
  const int sj = blockIdx.x;
  const int si = blockIdx.y;
  const int b = blockIdx.z;
  const int Pr = 4 * si, Pc = 4 * sj;          // padded-grid coords
  const int r = min(Pr, HPMAX), c = min(Pc, HPMAX);  // actual site coords
  const float* img = meanimg + (long)b * HW;

  // stage window rows [r-18, r+24] x cols [c-18, c+24] into LDS (clamped)
  // via the direct global->LDS async path.
  for (int e = tid; e < WINEXT * WINEXT; e += 64) {
    int wr = e / WINEXT, wc = e % WINEXT;
    int gr = min(max(r - V_HALF + wr, 0), H_IMG - 1);
    int gc = min(max(c - V_HALF + wc, 0), W_IMG - 1);
    async_g2lds_b32(&win[e], &img[gr * W_IMG + gc]);
  }
  wait_async0();
  __syncthreads();

  // distances for all 1369 offsets
  for (int o = tid; o < WIN_SZ * WIN_SZ; o += 64) {
    int oi = o / WIN_SZ, oj = o % WIN_SZ;
    float d;
    if (oi == V_HALF && oj == V_HALF) {
      d = -3.0e38f;  // center forced first
    } else {
      int qr = Pr + oi - V_HALF, qc = Pc + oj - V_HALF;
      int ar, ac;
      if (!mapq(qr, qc, ar, ac)) {
        d = 3.0e38f;
      } else {
        int br = ar - (r - V_HALF), bc = ac - (c - V_HALF);
        float s = 0.0f;
        for (int pi = 0; pi < P_PATCH; ++pi)
          for (int pj = 0; pj < P_PATCH; ++pj) {
            float dv = win[(br + pi) * WINEXT + bc + pj] -
                       win[(V_HALF + pi) * WINEXT + V_HALF + pj];
            s += dv * dv;
          }
        d = s * (1.0f / 49.0f);
      }
    }
    dist[o] = d;
  }
  __syncthreads();

  int* outp = idxout + ((long)b * NSITE * NSITE + si * NSITE + sj) * M;
  for (int k = 0; k < M; ++k) {
    float bv = 3.6e38f;
    int bo = WIN_SZ * WIN_SZ;
    for (int o = tid; o < WIN_SZ * WIN_SZ; o += 64) {
      float v = dist[o];
      if (v < bv) { bv = v; bo = o; }
    }
    rbest[tid] = bv;
    ribest[tid] = bo;
    __syncthreads();
    if (tid == 0) {
      float gv = rbest[0];
      int go = ribest[0];
      for (int t = 1; t < 64; ++t) {
        float v = rbest[t];
        int o = ribest[t];
        if (v < gv || (v == gv && o < go)) { gv = v; go = o; }
      }
      int oi = go / WIN_SZ, oj = go % WIN_SZ;
      // reference index math: ir = min((q - P) + r, H - p)
      int ir = min(Pr + oi - V_HALF - Pr + r, HPMAX);
      int ic = min(Pc + oj - V_HALF - Pc + c, HPMAX);
      outp[k] = ir * HOWO + ic;
      dist[go] = 3.5e38f;  // remove
    }
    __syncthreads();
  }
}

// ---------------------------------------------------------------------------
// Fused group denoise: async gather -> G = S S^T (WMMA) -> Cholesky ->
// theta = I - c A^{-1} -> X_hat = theta Y (WMMA) -> atomic aggregation.
// One workgroup (128 threads, 4 waves) per group; everything in LDS.
// ---------------------------------------------------------------------------
template <int M, int MP, bool STEP2>
__global__ void __launch_bounds__(128) denoise_kernel(
    const float* __restrict__ yimg, const float* __restrict__ ximg,
    const int* __restrict__ idxbuf, const float* __restrict__ sigma_p,
    float* __restrict__ xsum, float* __restrict__ wmap) {
  extern __shared__ float smem[];
  float* Ys = smem;                                        // MP x NPAD
  float* Xs = STEP2 ? (smem + MP * NPAD) : smem;           // MP x NPAD (step2)
  float* A = smem + (STEP2 ? 2 : 1) * MP * NPAD;           // MP x MP
  float* T = A + MP * MP;                                  // MP x MP
  float* wv = T + MP * MP;                                 // MP
  int* idxs = (int*)(wv + MP);                             // M

  const int tid = threadIdx.x;
  const long grp = blockIdx.x;
  const int b = (int)(grp / (NSITE * NSITE));
  const int* myidx = idxbuf + grp * M;
  if (tid < M) idxs[tid] = myidx[tid];
  __syncthreads();

  // gather Y (and X), zero-padded to MP x NPAD. Valid elements stream
  // straight L2 -> LDS (async); pad elements are zero-filled with ds_store.
  for (int e = tid; e < MP * NPAD; e += 128) {
    int k = e / NPAD, f = e % NPAD;
    if (k < M && f < NFEAT) {
      int q = idxs[k];
      int pr = q / HOWO, pc = q % HOWO;
      int cc = f / 49, rem = f % 49;
      int pi = rem / 7, pj = rem % 7;
      long off = ((long)b * C_IMG + cc) * HW + (pr + pi) * W_IMG + (pc + pj);
      async_g2lds_b32(&Ys[e], yimg + off);
      if (STEP2) async_g2lds_b32(&Xs[e], ximg + off);
    } else {
      Ys[e] = 0.0f;
      if (STEP2) Xs[e] = 0.0f;
    }
  }
  wait_async0();
  __syncthreads();

  const int lane = tid & 31;
  const int wid = tid >> 5;  // 4 waves
  const int hi = lane >> 4;  // half-wave select
  const int lo = lane & 15;
  const float* S = STEP2 ? Xs : Ys;
  const int TB = MP / 16;

  // G = S * S^T  (f32 WMMA 16x16x4, K chunks over 147 features)
  for (int t = wid; t < TB * TB; t += 4) {
    int bi = t / TB, bj = t % TB;
    v8f acc = {0.f, 0.f, 0.f, 0.f, 0.f, 0.f, 0.f, 0.f};
    for (int kc = 0; kc < 37; ++kc) {
      int k0 = kc * 4 + 2 * hi;
      const float* pa = S + (bi * 16 + lo) * NPAD + k0;
      const float* pb = S + (bj * 16 + lo) * NPAD + k0;
      v2f a = {pa[0], pa[1]};
      v2f bb = {pb[0], pb[1]};
      acc = wmma_f32x4(a, bb, acc);
    }
#pragma unroll
    for (int vg = 0; vg < 8; ++vg) {
      int row = bi * 16 + vg + 8 * hi;
      int col = bj * 16 + lo;
      A[row * MP + col] = acc[vg];
    }
  }
  __syncthreads();

  const float sg = *sigma_p;
  const float creg = (float)NFEAT * sg * sg;  // n * sigma^2
  if (STEP2) {
    if (tid < M) A[tid * MP + tid] += creg;
    __syncthreads();
  }

  // in-place Cholesky of A (m x m, lower)
  for (int k = 0; k < M; ++k) {
    if (tid == 0) A[k * MP + k] = sqrtf(A[k * MP + k]);
    __syncthreads();
    float dk = A[k * MP + k];
    for (int i = k + 1 + tid; i < M; i += 128) A[i * MP + k] /= dk;
    __syncthreads();
    int rem = M - k - 1;
    for (int e = tid; e < rem * rem; e += 128) {
      int i = k + 1 + e / rem, j = k + 1 + e % rem;
      if (i >= j) A[i * MP + j] -= A[i * MP + k] * A[j * MP + k];
    }
    __syncthreads();
  }

  // theta = I - c * A^{-1} : column j solved by thread j
  for (int e = tid; e < MP * MP; e += 128) T[e] = 0.0f;
  __syncthreads();
  if (tid < M) {
    int j = tid;
    T[j * MP + j] = 1.0f;
    for (int i = 0; i < M; ++i) {  // L y = e_j
      float s = T[i * MP + j];
      for (int t2 = 0; t2 < i; ++t2) s -= A[i * MP + t2] * T[t2 * MP + j];
      T[i * MP + j] = s / A[i * MP + i];
    }
    for (int i = M - 1; i >= 0; --i) {  // L^T x = y
      float s = T[i * MP + j];
      for (int t2 = i + 1; t2 < M; ++t2) s -= A[t2 * MP + i] * T[t2 * MP + j];
      T[i * MP + j] = s / A[i * MP + i];
    }
    for (int i = 0; i < M; ++i) {
      float mv = T[i * MP + j];
      T[i * MP + j] = ((i == j) ? 1.0f : 0.0f) - creg * mv;
    }
  }
  __syncthreads();

  // per-row weights
  if (tid < M) {
    float s = 0.0f;
    for (int j = 0; j < M; ++j) {
      float v = T[tid * MP + j];
      s += v * v;
    }
    s = fminf(fmaxf(s, 1.0f / (float)M), 1.0f);
    wv[tid] = 1.0f / s;
  }
  __syncthreads();

  // X_hat = theta * Y (WMMA), fused aggregation scatter
  float* xs_b = xsum + (long)b * C_IMG * HW;
  float* wm_b = wmap + (long)b * HW;
  for (int t = wid; t < TB * 10; t += 4) {
    int bi = t / 10, bj = t % 10;
    v8f acc = {0.f, 0.f, 0.f, 0.f, 0.f, 0.f, 0.f, 0.f};
    for (int kc = 0; kc < MP / 4; ++kc) {
      int k0 = kc * 4 + 2 * hi;
      v2f a = {T[(bi * 16 + lo) * MP + k0], T[(bi * 16 + lo) * MP + k0 + 1]};
      v2f bb = {Ys[k0 * NPAD + bj * 16 + lo], Ys[(k0 + 1) * NPAD + bj * 16 + lo]};
      acc = wmma_f32x4(a, bb, acc);
    }
#pragma unroll
    for (int vg = 0; vg < 8; ++vg) {
      int k = bi * 16 + vg + 8 * hi;
      int f = bj * 16 + lo;
      if (k < M && f < NFEAT) {
        int q = idxs[k];
        int pr = q / HOWO, pc = q % HOWO;
        int cc = f / 49, rem = f % 49;
        int pi = rem / 7, pj = rem % 7;
        atomicAdd(&xs_b[cc * HW + (pr + pi) * W_IMG + (pc + pj)],
                  acc[vg] * wv[k]);
      }
    }
  }
  // weight map
  for (int e = tid; e < M * 49; e += 128) {
    int k = e / 49, rem = e % 49;
    int pi = rem / 7, pj = rem % 7;
    int q = idxs[k];
    int pr = q / HOWO, pc = q % HOWO;
    atomicAdd(&wm_b[(pr + pi) * W_IMG + (pc + pj)], wv[k]);
  }
}

// ---------------------------------------------------------------------------
extern "C" void kernel_launch(void* const* d_in, const int* in_sizes, int n_in,
                              void* d_out, int out_size, void* d_ws,
                              size_t ws_size, hipStream_t stream) {
  const float* y = (const float*)d_in[0];
  const float* sigma = (const float*)d_in[1];
  float* out = (float*)d_out;

  float* ws = (float*)d_ws;
  float* meanb = ws;                    // N*HW
  float* xsum = meanb + N_IMG * HW;     // N*C*HW
  float* wmap = xsum + N_IMG * C_IMG * HW;  // N*HW (adjacent to xsum)
  float* den1 = wmap + N_IMG * HW;      // N*C*HW
  int* idx1 = (int*)(den1 + N_IMG * C_IMG * HW);  // N*4096*18
  int* idx2 = idx1 + N_IMG * NSITE * NSITE * 18;  // N*4096*55

  const int nMean = N_IMG * HW;
  const int nOut = N_IMG * C_IMG * HW;
  const int nAcc = nOut + nMean;  // xsum + wmap contiguous
  const int nGroups = N_IMG * NSITE * NSITE;

  const size_t lds1 = (size_t)(1 * 32 * NPAD + 2 * 32 * 32 + 32) * 4 + 18 * 4;
  const size_t lds2 = (size_t)(2 * 64 * NPAD + 2 * 64 * 64 + 64) * 4 + 55 * 4;

  // ---- step 1 ----
  mean_kernel<<<(nMean + 255) / 256, 256, 0, stream>>>(y, meanb, nMean);
  zero_kernel<<<(nAcc + 255) / 256, 256, 0, stream>>>(xsum, nAcc);
  bm_kernel<18><<<dim3(NSITE, NSITE, N_IMG), 64, 0, stream>>>(meanb, idx1);
  denoise_kernel<18, 32, false>
      <<<nGroups, 128, lds1, stream>>>(y, y, idx1, sigma, xsum, wmap);
  divide_kernel<<<(nOut + 255) / 256, 256, 0, stream>>>(xsum, wmap, den1, nOut);

  // ---- step 2 ----
  mean_kernel<<<(nMean + 255) / 256, 256, 0, stream>>>(den1, meanb, nMean);
  zero_kernel<<<(nAcc + 255) / 256, 256, 0, stream>>>(xsum, nAcc);
  bm_kernel<55><<<dim3(NSITE, NSITE, N_IMG), 64, 0, stream>>>(meanb, idx2);
  denoise_kernel<55, 64, true>
      <<<nGroups, 128, lds2, stream>>>(y, den1, idx2, sigma, xsum, wmap);
  divide_kernel<<<(nOut + 255) / 256, 256, 0, stream>>>(xsum, wmap, out, nOut);
}